// OptimizedVoxelizer_57062935495128
// MI455X (gfx1250) — compile-verified
//
#include <hip/hip_runtime.h>
#include <hip/hip_bf16.h>

typedef __attribute__((ext_vector_type(2))) float v2f;
typedef __attribute__((ext_vector_type(8))) float v8f;

#define GX 200
#define GY 200
#define GZ 16
#define NVOX (GX * GY * GZ)
#define NCH 16
#define OPA_THRESH 1e-4f
#define EPSV 1e-6f

__global__ void vox_zero_kernel(float* __restrict__ p, int n) {
    int i = blockIdx.x * blockDim.x + threadIdx.x;
    if (i < n) p[i] = 0.0f;
}

// Flat voxel index for voxel `vox` of this Gaussian's 8^3 window.
// ix0/iy0/iz0 are wave-uniform and >= 0; only the upper clamp is needed
// (matches reference's clip of idx to gshape-1).
__device__ __forceinline__ int flat_of(int vox, int ix0, int iy0, int iz0) {
    const int ox = vox >> 6;
    const int oy = (vox >> 3) & 7;
    const int oz = vox & 7;
    const int jx = min(ix0 + ox, GX - 1);
    const int jy = min(iy0 + oy, GY - 1);
    const int jz = min(iz0 + oz, GZ - 1);
    return (jx * GY + jy) * GZ + jz;
}

// One wave (32 lanes) per Gaussian. 512-voxel window processed as chunks of
// 32 voxels; chunk loop trimmed (wave-uniform) to the Gaussian's 3-sigma bbox.
// Feature outer-product (contrib x features) via two v_wmma_f32_16x16x4_f32:
//   A (16x4 f32 layout): a.x lanes 0-15 = column K=0 (voxels 0-15),
//                        a.x lanes 16-31 = column K=2 (voxels 16-31).
//   B1: features in row K=0 (b.x, lanes 0-15)  -> D1 = contrib[0:16]  (x) feat
//   B2: features in row K=2 (b.x, lanes 16-31) -> D2 = contrib[16:32] (x) feat
// No cross-lane data movement needed anywhere.
__global__ void __launch_bounds__(256)
vox_splat_kernel(const float* __restrict__ means,
                 const float* __restrict__ opac,
                 const float* __restrict__ cov,
                 const float* __restrict__ feat,
                 float* __restrict__ density,
                 float* __restrict__ feats,
                 int N) {
    const int lane = threadIdx.x & 31;
    const int g = blockIdx.x * (blockDim.x >> 5) + (threadIdx.x >> 5);
    if (g >= N) return;  // wave-uniform

    const float vminx = -50.f, vminy = -50.f, vminz = -5.f;
    const float vmaxx = 50.f, vmaxy = 50.f, vmaxz = 3.f;

    // ---- per-Gaussian setup (redundant across lanes; L0/L2 served) ----
    const float mx = means[g * 3 + 0];
    const float my = means[g * 3 + 1];
    const float mz = means[g * 3 + 2];
    const float op = opac[g];
    const float c00 = cov[g * 9 + 0], c01 = cov[g * 9 + 1], c02 = cov[g * 9 + 2];
    const float c11 = cov[g * 9 + 4], c12 = cov[g * 9 + 5], c22 = cov[g * 9 + 8];

    // inverse of symmetric 3x3 (adjugate / det)
    const float A0 = c11 * c22 - c12 * c12;
    const float B0 = c02 * c12 - c01 * c22;
    const float C0 = c01 * c12 - c02 * c11;
    const float rdet = 1.0f / (c00 * A0 + c01 * B0 + c02 * C0);
    const float i00 = A0 * rdet;
    const float i01 = B0 * rdet;
    const float i02 = C0 * rdet;
    const float i11 = (c00 * c22 - c02 * c02) * rdet;
    const float i12 = (c01 * c02 - c00 * c12) * rdet;
    const float i22 = (c00 * c11 - c01 * c01) * rdet;

    // 3-sigma bbox + prefilter
    const float sx = 3.0f * sqrtf(c00);
    const float sy = 3.0f * sqrtf(c11);
    const float sz = 3.0f * sqrtf(c22);
    const float bminx = mx - sx, bminy = my - sy, bminz = mz - sz;
    const float bmaxx = mx + sx, bmaxy = my + sy, bmaxz = mz + sz;
    const bool fmask = (bmaxx > vminx) && (bmaxy > vminy) && (bmaxz > vminz) &&
                       (bminx < vmaxx) && (bminy < vmaxy) && (bminz < vmaxz) &&
                       (op > OPA_THRESH);

    const float blx = fminf(fmaxf(bminx, vminx), vmaxx);
    const float bly = fminf(fmaxf(bminy, vminy), vmaxy);
    const float blz = fminf(fmaxf(bminz, vminz), vmaxz);
    const float bhx = fminf(fmaxf(bmaxx, vminx), vmaxx);
    const float bhy = fminf(fmaxf(bmaxy, vminy), vmaxy);
    const float bhz = fminf(fmaxf(bmaxz, vminz), vmaxz);
    const int ix0 = max((int)((blx - vminx) * 2.0f), 0);
    const int iy0 = max((int)((bly - vminy) * 2.0f), 0);
    const int iz0 = max((int)((blz - vminz) * 2.0f), 0);
    const int ix1 = min((int)((bhx - vminx) * 2.0f), GX - 1);
    const int iy1 = min((int)((bhy - vminy) * 2.0f), GY - 1);
    const int iz1 = min((int)((bhz - vminz) * 2.0f), GZ - 1);
    const int dxm = ix1 - ix0, dym = iy1 - iy0, dzm = iz1 - iz0;

    // Wave-uniform chunk-loop trimming:
    //   chunk c covers ox = c>>1, oy in {0..3} (even c) / {4..7} (odd c).
    const int cmax = fmask ? min(16, 2 * (dxm + 1)) : 0;
    const int cstep = (dym <= 3) ? 2 : 1;  // skip oy>=4 chunks if y-span <= 4

    // B matrices (hoisted; constant per Gaussian).
    const int nch = lane & 15;
    const float fv = feat[g * NCH + nch];
    v2f bmat1, bmat2;
    bmat1.x = (lane < 16) ? fv : 0.0f;  // row K=0 = features
    bmat1.y = 0.0f;
    bmat2.x = (lane < 16) ? 0.0f : fv;  // row K=2 = features
    bmat2.y = 0.0f;

    const v8f accz = {};               // zero accumulator
    const int hi = (lane >> 4) << 3;   // D layout: M = r + 8*(lane>=16)

    for (int c = 0; c < cmax; c += cstep) {
        const int base = c * 32;
        const int vox = base + lane;
        const int ox = vox >> 6;
        const int oy = (vox >> 3) & 7;
        const int oz = vox & 7;
        const bool valid = (ox <= dxm) && (oy <= dym) && (oz <= dzm);

        // wave-uniform skip before any expensive math
        if (__ballot(valid) == 0ull) continue;

        float contrib = 0.0f;
        if (valid) {
            const float ddx = ((float)(ix0 + ox) * 0.5f + vminx + 0.25f) - mx;
            const float ddy = ((float)(iy0 + oy) * 0.5f + vminy + 0.25f) - my;
            const float ddz = ((float)(iz0 + oz) * 0.5f + vminz + 0.25f) - mz;
            const float maha =
                i00 * ddx * ddx + i11 * ddy * ddy + i22 * ddz * ddz +
                2.0f * (i01 * ddx * ddy + i02 * ddx * ddz + i12 * ddy * ddz);
            contrib = op * __expf(-0.5f * maha);
        }

        if (contrib != 0.0f) {
            const int flat = flat_of(vox, ix0, iy0, iz0);
            atomicAdd(&density[flat], contrib);
        }

        // A matrix: contrib in a.x for all 32 lanes
        //   -> column K=0 holds voxels base+0..15, column K=2 holds base+16..31.
        v2f amat;
        amat.x = contrib;
        amat.y = 0.0f;

        const v8f d1 = __builtin_amdgcn_wmma_f32_16x16x4_f32(
            false, amat, false, bmat1, (short)0, accz, false, false);
        const v8f d2 = __builtin_amdgcn_wmma_f32_16x16x4_f32(
            false, amat, false, bmat2, (short)0, accz, false, false);

#pragma unroll
        for (int r = 0; r < 8; ++r) {
            const int mloc = r + hi;  // this lane's D-row voxel within half-chunk
            const float w1 = d1[r];
            if (w1 != 0.0f) {
                const int fl = flat_of(base + mloc, ix0, iy0, iz0);
                atomicAdd(&feats[fl * NCH + nch], w1);
            }
            const float w2 = d2[r];
            if (w2 != 0.0f) {
                const int fl = flat_of(base + 16 + mloc, ix0, iy0, iz0);
                atomicAdd(&feats[fl * NCH + nch], w2);
            }
        }
    }
}

__global__ void vox_norm_kernel(const float* __restrict__ density,
                                float* __restrict__ feats) {
    int v = blockIdx.x * blockDim.x + threadIdx.x;
    if (v >= NVOX) return;
    const float s = 1.0f / fmaxf(density[v], EPSV);
    float4* fp = (float4*)(feats + (size_t)v * NCH);
#pragma unroll
    for (int i = 0; i < 4; ++i) {
        float4 f = fp[i];
        f.x *= s; f.y *= s; f.z *= s; f.w *= s;
        fp[i] = f;
    }
}

extern "C" void kernel_launch(void* const* d_in, const int* in_sizes, int n_in,
                              void* d_out, int out_size, void* d_ws, size_t ws_size,
                              hipStream_t stream) {
    const float* means = (const float*)d_in[0];  // [N,3]
    const float* opac  = (const float*)d_in[1];  // [N,1]
    const float* cov   = (const float*)d_in[2];  // [N,3,3]
    const float* feat  = (const float*)d_in[3];  // [N,16]
    const int N = in_sizes[0] / 3;

    float* density = (float*)d_out;      // [200*200*16]
    float* feats   = density + NVOX;     // [200*200*16, 16]

    const int total = NVOX * (1 + NCH);
    vox_zero_kernel<<<(total + 255) / 256, 256, 0, stream>>>(density, total);

    const int waves_per_block = 8;  // 256 threads = 8 wave32
    const int blocks = (N + waves_per_block - 1) / waves_per_block;
    vox_splat_kernel<<<blocks, 256, 0, stream>>>(means, opac, cov, feat,
                                                 density, feats, N);

    vox_norm_kernel<<<(NVOX + 255) / 256, 256, 0, stream>>>(density, feats);
}